// HeteroGATLayer_61280593379539
// MI455X (gfx1250) — compile-verified
//
#include <hip/hip_runtime.h>
#include <math.h>

typedef __attribute__((ext_vector_type(2))) float v2f;
typedef __attribute__((ext_vector_type(8))) float v8f;

constexpr int NOP   = 50000;
constexpr int NMAC  = 2000;
constexpr int OUTD  = 128;
constexpr int HEADS = 4;
constexpr int DK    = 32;     // OUTD / HEADS
constexpr int ESEQ  = 150000;
constexpr int EOM   = 300000;
constexpr int EMO   = 300000;

// ---------------------------------------------------------------------------
// GEMM + bias via V_WMMA_F32_16X16X4_F32.
// Z[M,128] = A[M,K] @ W[128,K]^T + bias.  One wave per 16x16 output tile.
// A-fragment (16x4 f32, 2 VGPRs): lane L holds row m=L&15, K=(k0+2*(L>>4))..+1
// B-fragment (4x16 f32, 2 VGPRs): lane L holds col n=L&15, same K pair.
//   Since B[k][n] = W[n][k], both fragments are contiguous 8-byte row loads.
// C/D (8 VGPRs): lane L holds n=L&15, VGPR r holds m = r + 8*(L>>4).
// ---------------------------------------------------------------------------
template <int K>
__global__ void gemm_bias_wmma(const float* __restrict__ A,
                               const float* __restrict__ W,
                               const float* __restrict__ bias,
                               float* __restrict__ Z, int M) {
  const int lane  = threadIdx.x & 31;
  const int wave  = blockIdx.x * (blockDim.x >> 5) + (threadIdx.x >> 5);
  const int ntiles = OUTD / 16;                 // 8
  const int mtile  = wave / ntiles;
  const int ntile  = wave - mtile * ntiles;
  if (mtile * 16 >= M) return;                  // whole-wave uniform exit

  const int half = lane >> 4;
  const int l15  = lane & 15;
  const float* __restrict__ arow = A + (size_t)(mtile * 16 + l15) * K;
  const float* __restrict__ wrow = W + (size_t)(ntile * 16 + l15) * K;

  v8f acc = {};
#pragma unroll
  for (int k0 = 0; k0 < K; k0 += 4) {
    v2f a = *(const v2f*)(arow + k0 + 2 * half);
    v2f b = *(const v2f*)(wrow + k0 + 2 * half);
    acc = __builtin_amdgcn_wmma_f32_16x16x4_f32(
        /*neg_a=*/false, a, /*neg_b=*/false, b,
        /*c_mod=*/(short)0, acc, /*reuse_a=*/false, /*reuse_b=*/false);
  }

  const int   n  = ntile * 16 + l15;
  const float bv = bias[n];
  float* __restrict__ zc = Z + n;
#pragma unroll
  for (int r = 0; r < 8; ++r) {
    const int m = mtile * 16 + r + 8 * half;
    zc[(size_t)m * OUTD] = acc[r] + bv;
  }
}

// ---------------------------------------------------------------------------
// Pass 1: per-(edge,head) attention score -> alpha, and denom[dst,h] += alpha
// score = dot(h_s, att[0:32]) + dot(h_d, att[32:64]) + feat*att[64]
// ---------------------------------------------------------------------------
__global__ void edge_score(const float* __restrict__ Zs,
                           const float* __restrict__ Zd,
                           const int* __restrict__ src,
                           const int* __restrict__ dst,
                           const float* __restrict__ feat,
                           const float* __restrict__ att,      // [4, 2*DK+1]
                           float* __restrict__ alpha,          // [E, 4]
                           float* __restrict__ denom,          // [Nd, 4]
                           int E) {
  const int t = blockIdx.x * blockDim.x + threadIdx.x;
  const int e = t >> 2;
  if (e >= E) return;
  const int h = t & 3;

  const int s = src[e];
  const int d = dst[e];
  const float* __restrict__ hs = Zs + (size_t)s * OUTD + h * DK;
  const float* __restrict__ hd = Zd + (size_t)d * OUTD + h * DK;
  const float* __restrict__ a  = att + h * (2 * DK + 1);

  float sc = feat[e] * a[2 * DK];
#pragma unroll
  for (int i = 0; i < DK; i += 4) {
    const float4 x = *(const float4*)(hs + i);
    const float4 y = *(const float4*)(hd + i);
    sc += x.x * a[i]          + x.y * a[i + 1]
        + x.z * a[i + 2]      + x.w * a[i + 3];
    sc += y.x * a[DK + i]     + y.y * a[DK + i + 1]
        + y.z * a[DK + i + 2] + y.w * a[DK + i + 3];
  }
  sc = (sc > 0.0f) ? sc : 0.2f * sc;          // leaky_relu(0.2)
  sc = fminf(fmaxf(sc, -20.0f), 20.0f);       // clip
  const float al = expf(sc);
  alpha[(size_t)e * 4 + h] = al;
  atomicAdd(denom + (size_t)d * 4 + h, al);
}

// ---------------------------------------------------------------------------
// Pass 2: per-(edge,channel) weighted scatter:  agg[dst,c] += h_s[c]*alpha/den
// ---------------------------------------------------------------------------
__global__ void edge_aggregate(const float* __restrict__ Zs,
                               const int* __restrict__ src,
                               const int* __restrict__ dst,
                               const float* __restrict__ alpha,
                               const float* __restrict__ denom,
                               float* __restrict__ agg, int E) {
  const int t = blockIdx.x * blockDim.x + threadIdx.x;
  const int e = t >> 7;
  if (e >= E) return;
  const int c = t & 127;
  const int h = c >> 5;
  const int s = src[e];
  const int d = dst[e];
  const float w = alpha[(size_t)e * 4 + h] / (denom[(size_t)d * 4 + h] + 1e-6f);
  atomicAdd(agg + (size_t)d * OUTD + c, Zs[(size_t)s * OUTD + c] * w);
}

// ---------------------------------------------------------------------------
// Finalize: res = agg + z; LayerNorm(128); ELU. One wave32 per row,
// each lane owns 4 contiguous channels; shuffle reductions (wave32-safe).
// ---------------------------------------------------------------------------
__global__ void finalize_ln_elu(const float* __restrict__ agg,
                                const float* __restrict__ z,
                                const float* __restrict__ g,
                                const float* __restrict__ b,
                                float* __restrict__ out, int N) {
  const int row  = blockIdx.x * (blockDim.x >> 5) + (threadIdx.x >> 5);
  const int lane = threadIdx.x & 31;
  if (row >= N) return;
  const int c = lane * 4;

  const float4 av = *(const float4*)(agg + (size_t)row * OUTD + c);
  const float4 zv = *(const float4*)(z   + (size_t)row * OUTD + c);
  float x0 = av.x + zv.x, x1 = av.y + zv.y, x2 = av.z + zv.z, x3 = av.w + zv.w;

  float s = x0 + x1 + x2 + x3;
#pragma unroll
  for (int m = 16; m >= 1; m >>= 1) s += __shfl_xor(s, m, 32);
  const float mu = s * (1.0f / OUTD);

  const float d0 = x0 - mu, d1 = x1 - mu, d2 = x2 - mu, d3 = x3 - mu;
  float v = d0 * d0 + d1 * d1 + d2 * d2 + d3 * d3;
#pragma unroll
  for (int m = 16; m >= 1; m >>= 1) v += __shfl_xor(v, m, 32);
  const float rstd = rsqrtf(v * (1.0f / OUTD) + 1e-5f);

  const float4 gv = *(const float4*)(g + c);
  const float4 bv = *(const float4*)(b + c);
  float y0 = d0 * rstd * gv.x + bv.x;
  float y1 = d1 * rstd * gv.y + bv.y;
  float y2 = d2 * rstd * gv.z + bv.z;
  float y3 = d3 * rstd * gv.w + bv.w;
  y0 = (y0 > 0.0f) ? y0 : expf(y0) - 1.0f;    // ELU
  y1 = (y1 > 0.0f) ? y1 : expf(y1) - 1.0f;
  y2 = (y2 > 0.0f) ? y2 : expf(y2) - 1.0f;
  y3 = (y3 > 0.0f) ? y3 : expf(y3) - 1.0f;
  float4 yo = {y0, y1, y2, y3};
  *(float4*)(out + (size_t)row * OUTD + c) = yo;
}

extern "C" void kernel_launch(void* const* d_in, const int* in_sizes, int n_in,
                              void* d_out, int out_size, void* d_ws, size_t ws_size,
                              hipStream_t stream) {
  const float* h_op     = (const float*)d_in[0];
  const float* h_mac    = (const float*)d_in[1];
  const int*   seq_src  = (const int*)d_in[2];
  const int*   seq_dst  = (const int*)d_in[3];
  const int*   om_src   = (const int*)d_in[4];
  const int*   om_dst   = (const int*)d_in[5];
  const int*   mo_src   = (const int*)d_in[6];
  const int*   mo_dst   = (const int*)d_in[7];
  const float* feat_seq = (const float*)d_in[8];
  const float* feat_om  = (const float*)d_in[9];
  const float* feat_mo  = (const float*)d_in[10];
  const float* W_op_w   = (const float*)d_in[11];
  const float* W_op_b   = (const float*)d_in[12];
  const float* W_mac_w  = (const float*)d_in[13];
  const float* W_mac_b  = (const float*)d_in[14];
  const float* att_seq  = (const float*)d_in[15];
  const float* att_om   = (const float*)d_in[16];
  const float* att_mo   = (const float*)d_in[17];
  const float* ln_op_g  = (const float*)d_in[18];
  const float* ln_op_b  = (const float*)d_in[19];
  const float* ln_mac_g = (const float*)d_in[20];
  const float* ln_mac_b = (const float*)d_in[21];

  // Workspace layout (floats). Regions that must start at zero are contiguous
  // and first, so a single async memset clears them (graph-capture safe).
  float* ws        = (float*)d_ws;
  float* agg_op    = ws;                                     // NOP*OUTD
  float* agg_mac   = agg_op    + (size_t)NOP * OUTD;         // NMAC*OUTD
  float* denom_seq = agg_mac   + (size_t)NMAC * OUTD;        // NOP*HEADS
  float* denom_mo  = denom_seq + (size_t)NOP * HEADS;        // NOP*HEADS
  float* denom_om  = denom_mo  + (size_t)NOP * HEADS;        // NMAC*HEADS
  float* z_op      = denom_om  + (size_t)NMAC * HEADS;       // NOP*OUTD
  float* z_mac     = z_op      + (size_t)NOP * OUTD;         // NMAC*OUTD
  float* alpha_seq = z_mac     + (size_t)NMAC * OUTD;        // ESEQ*HEADS
  float* alpha_mo  = alpha_seq + (size_t)ESEQ * HEADS;       // EMO*HEADS
  float* alpha_om  = alpha_mo  + (size_t)EMO * HEADS;        // EOM*HEADS

  const size_t zero_bytes = (size_t)(z_op - agg_op) * sizeof(float);
  hipMemsetAsync(agg_op, 0, zero_bytes, stream);

  // Dense projections (WMMA f32 16x16x4)
  {
    const int waves = (NOP / 16) * (OUTD / 16);
    gemm_bias_wmma<64><<<(waves + 7) / 8, 256, 0, stream>>>(h_op, W_op_w, W_op_b, z_op, NOP);
  }
  {
    const int waves = (NMAC / 16) * (OUTD / 16);
    gemm_bias_wmma<32><<<(waves + 7) / 8, 256, 0, stream>>>(h_mac, W_mac_w, W_mac_b, z_mac, NMAC);
  }

  // Attention scores + softmax denominators
  edge_score<<<(ESEQ * HEADS + 255) / 256, 256, 0, stream>>>(
      z_op, z_op, seq_src, seq_dst, feat_seq, att_seq, alpha_seq, denom_seq, ESEQ);
  edge_score<<<(EMO * HEADS + 255) / 256, 256, 0, stream>>>(
      z_mac, z_op, mo_src, mo_dst, feat_mo, att_mo, alpha_mo, denom_mo, EMO);
  edge_score<<<(EOM * HEADS + 255) / 256, 256, 0, stream>>>(
      z_op, z_mac, om_src, om_dst, feat_om, att_om, alpha_om, denom_om, EOM);

  // Normalized weighted aggregation (both op-destined attns sum into agg_op)
  edge_aggregate<<<((size_t)ESEQ * OUTD + 255) / 256, 256, 0, stream>>>(
      z_op, seq_src, seq_dst, alpha_seq, denom_seq, agg_op, ESEQ);
  edge_aggregate<<<((size_t)EMO * OUTD + 255) / 256, 256, 0, stream>>>(
      z_mac, mo_src, mo_dst, alpha_mo, denom_mo, agg_op, EMO);
  edge_aggregate<<<((size_t)EOM * OUTD + 255) / 256, 256, 0, stream>>>(
      z_op, om_src, om_dst, alpha_om, denom_om, agg_mac, EOM);

  // Residual + LayerNorm + ELU
  float* out_op  = (float*)d_out;
  float* out_mac = out_op + (size_t)NOP * OUTD;
  finalize_ln_elu<<<(NOP + 7) / 8, 256, 0, stream>>>(agg_op, z_op, ln_op_g, ln_op_b, out_op, NOP);
  finalize_ln_elu<<<(NMAC + 7) / 8, 256, 0, stream>>>(agg_mac, z_mac, ln_mac_g, ln_mac_b, out_mac, NMAC);
}